// CenterLoss_807453851770
// MI455X (gfx1250) — compile-verified
//
#include <hip/hip_runtime.h>
#include <hip/hip_bf16.h>

// CenterLoss masked BCE-with-logits reduction for MI455X (gfx1250, wave32).
// Inputs:  d_in[0] pred   f32 [32,8,512,512]
//          d_in[1] target f32 [32,8,512,512]
//          d_in[2] rand_mask bool(1B) [32,1,512,512]
// Output:  d_out[0] f32 scalar loss
// Scratch: d_ws needs 2*1024 floats (8 KB): block partials for loss & weight.

typedef __attribute__((ext_vector_type(2))) float v2f;
typedef __attribute__((ext_vector_type(8))) float v8f;

#define NN 32
#define CC 8
#define HW 262144                 // 512*512
#define GROUPS_ (NN * HW / 4)     // 2,097,152 float4 pixel-groups
#define NBLOCKS 1024
#define NTHREADS 256

__device__ __forceinline__ float bce_term(float x, float t) {
  // max(x,0) - x*t + log1p(exp(-|x|)); e in (0,1] so log(1+e) is well conditioned
  float e = __expf(-__builtin_fabsf(x));
  float l = __logf(1.0f + e);
  return __builtin_fmaxf(x, 0.0f) - x * t + l;
}

__global__ __launch_bounds__(NTHREADS) void bce_partial_kernel(
    const float* __restrict__ pred, const float* __restrict__ target,
    const unsigned char* __restrict__ mask, float* __restrict__ part) {
  const int tid = blockIdx.x * NTHREADS + threadIdx.x;
  const int stride = NBLOCKS * NTHREADS;  // 262,144 threads -> 8 iters each
  float lsum = 0.0f, wsum = 0.0f;

  for (int g = tid; g < GROUPS_; g += stride) {
    const int n = g >> 16;                // g / (HW/4), HW/4 = 65536
    const int hw4 = (g & 0xFFFF) << 2;    // pixel offset within image
    const int pixbase = n * HW + hw4;
    const int cbase = n * (CC * HW) + hw4;

    // 4 mask bytes in one u32 (hw4 is a multiple of 4 -> aligned)
    const unsigned int mb =
        *reinterpret_cast<const unsigned int*>(mask + pixbase);

    // Load all 8 channel targets (float4 each) once; keep in registers.
    float4 tg[CC];
#pragma unroll
    for (int c = 0; c < CC; ++c)
      tg[c] = *reinterpret_cast<const float4*>(target + cbase + c * HW);

    // Speculative prefetch of channel 4 (+4*HW) of this pixel group (global_prefetch_b8).
    __builtin_prefetch(target + cbase + stride * 4, 0, 1);
    __builtin_prefetch(pred + cbase + stride * 4, 0, 1);

    float mx0 = tg[0].x, mx1 = tg[0].y, mx2 = tg[0].z, mx3 = tg[0].w;
#pragma unroll
    for (int c = 1; c < CC; ++c) {
      mx0 = __builtin_fmaxf(mx0, tg[c].x);
      mx1 = __builtin_fmaxf(mx1, tg[c].y);
      mx2 = __builtin_fmaxf(mx2, tg[c].z);
      mx3 = __builtin_fmaxf(mx3, tg[c].w);
    }
    const float w0 = ((mb & 0x000000FFu) != 0u || mx0 > 0.5f) ? 1.0f : 0.0f;
    const float w1 = ((mb & 0x0000FF00u) != 0u || mx1 > 0.5f) ? 1.0f : 0.0f;
    const float w2 = ((mb & 0x00FF0000u) != 0u || mx2 > 0.5f) ? 1.0f : 0.0f;
    const float w3 = ((mb & 0xFF000000u) != 0u || mx3 > 0.5f) ? 1.0f : 0.0f;
    const float wq = w0 + w1 + w2 + w3;
    wsum += wq;

    // Lanes whose whole quad has zero weight skip their pred cachelines
    // (EXEC-masked VMEM fetches only active lanes' lines).
    if (wq > 0.0f) {
#pragma unroll
      for (int c = 0; c < CC; ++c) {
        const float4 p = *reinterpret_cast<const float4*>(pred + cbase + c * HW);
        lsum += w0 * bce_term(p.x, tg[c].x);
        lsum += w1 * bce_term(p.y, tg[c].y);
        lsum += w2 * bce_term(p.z, tg[c].z);
        lsum += w3 * bce_term(p.w, tg[c].w);
      }
    }
  }

  // Deterministic block reduction: wave32 shuffle tree, then LDS across 8 waves.
#pragma unroll
  for (int off = 16; off > 0; off >>= 1) {
    lsum += __shfl_xor(lsum, off, 32);
    wsum += __shfl_xor(wsum, off, 32);
  }
  __shared__ float sl[NTHREADS / 32];
  __shared__ float sw[NTHREADS / 32];
  const int lane = threadIdx.x & 31;
  const int wid = threadIdx.x >> 5;
  if (lane == 0) { sl[wid] = lsum; sw[wid] = wsum; }
  __syncthreads();
  if (threadIdx.x == 0) {
    float L = 0.0f, Wt = 0.0f;
#pragma unroll
    for (int i = 0; i < NTHREADS / 32; ++i) { L += sl[i]; Wt += sw[i]; }
    part[blockIdx.x] = L;            // loss partial
    part[NBLOCKS + blockIdx.x] = Wt; // weight partial
  }
}

// Final reduction of 1024+1024 partials with the matrix engine:
// D = A(16x4 partial tile) x B(4x16 ones) + C, 16 accumulating WMMAs per sum.
// A f32 layout (ISA 7.12.2): lane L holds row M=L%16; VGPR v holds
// K = 2*(L>=16) + v  ->  flat index t*64 + M*4 + K.
// C f32 layout: VGPR j = row j (lanes 0-15) / row j+8 (lanes 16-31), so
// total = sum_j C[j]  +  same from opposite half-wave.
__global__ __launch_bounds__(32) void wmma_reduce_kernel(
    const float* __restrict__ part, float* __restrict__ out) {
  const int lane = threadIdx.x;       // one full wave32, EXEC all ones
  const int m = lane & 15;
  const int koff = (lane >> 4) << 1;  // 0 or 2

  v2f b;  b[0] = 1.0f; b[1] = 1.0f;   // ones B matrix (layout-invariant)
  v8f cl = {};
  v8f cw = {};

#pragma unroll
  for (int t = 0; t < 16; ++t) {
    const int idx = t * 64 + m * 4 + koff;          // even -> 8B aligned
    const float2 a0 = *reinterpret_cast<const float2*>(part + idx);
    const float2 a1 = *reinterpret_cast<const float2*>(part + NBLOCKS + idx);
    v2f al; al[0] = a0.x; al[1] = a0.y;
    v2f aw; aw[0] = a1.x; aw[1] = a1.y;
    cl = __builtin_amdgcn_wmma_f32_16x16x4_f32(false, al, false, b, (short)0,
                                               cl, false, false);
    cw = __builtin_amdgcn_wmma_f32_16x16x4_f32(false, aw, false, b, (short)0,
                                               cw, false, false);
  }

  float rl = 0.0f, rw = 0.0f;
#pragma unroll
  for (int j = 0; j < 8; ++j) { rl += cl[j]; rw += cw[j]; }
  rl += __shfl_xor(rl, 16, 32);   // combine rows 0-7 with rows 8-15
  rw += __shfl_xor(rw, 16, 32);

  if (lane == 0) out[0] = rl / (rw * (float)CC);
}

extern "C" void kernel_launch(void* const* d_in, const int* in_sizes, int n_in,
                              void* d_out, int out_size, void* d_ws,
                              size_t ws_size, hipStream_t stream) {
  (void)in_sizes; (void)n_in; (void)out_size; (void)ws_size;
  const float* pred = (const float*)d_in[0];
  const float* target = (const float*)d_in[1];
  const unsigned char* mask = (const unsigned char*)d_in[2];
  float* part = (float*)d_ws;   // 2048 floats used
  float* out = (float*)d_out;

  bce_partial_kernel<<<NBLOCKS, NTHREADS, 0, stream>>>(pred, target, mask, part);
  wmma_reduce_kernel<<<1, 32, 0, stream>>>(part, out);
}